// QuGCN_6794638263034
// MI455X (gfx1250) — compile-verified
//
#include <hip/hip_runtime.h>
#include <math.h>

// ---------------------------------------------------------------------------
// QuGCN chain-product on MI455X (gfx1250, wave32).
//
//   M_e = U rho_e U^dag = w_e w_e^dag  (rank-1, w_e = U c_e / ||c_e||)
//   chain = w_0 * prod_i <w_i, w_{i+1}> * w_{E-1}^dag
//
// Kernel 1: per-block (1024 edges) WMMA f32 16x16x4 batched matvec U*c
//           into LDS, then neighbor inner products + complex product tree.
// Kernel 2: 512-way complex product, outer product, linear, softmax.
// ---------------------------------------------------------------------------

#define N_EDGES   524288
#define CHUNK     1024
#define NBLOCKS   (N_EDGES / CHUNK)     // 512
#define NTHREADS  256                   // 8 wave32 waves
#define W_MUL     0.63245553203367586640f   // sqrt(2)/sqrt(5)

// ws layout (floats):
//  [0 .. 2*NBLOCKS)       per-block complex partial products (re, im)
//  [WS_W0 .. +8)          w_hat(edge 0):   re[0..3], im[0..3]
//  [WS_WL .. +8)          w_hat(edge E-1): re[0..3], im[0..3]
#define WS_W0   (2 * NBLOCKS)
#define WS_WL   (2 * NBLOCKS + 8)

typedef float v2f __attribute__((ext_vector_type(2)));
typedef float v8f __attribute__((ext_vector_type(8)));

struct C { float r, i; };
__device__ inline C cmul(C a, C b) { return { a.r*b.r - a.i*b.i, a.r*b.i + a.i*b.r }; }
__device__ inline C cadd(C a, C b) { return { a.r + b.r, a.i + b.i }; }

__device__ inline void mm2(const C* A, const C* B, C* D) {
    #pragma unroll
    for (int i = 0; i < 2; ++i)
        #pragma unroll
        for (int j = 0; j < 2; ++j)
            D[i*2 + j] = cadd(cmul(A[i*2 + 0], B[0*2 + j]),
                              cmul(A[i*2 + 1], B[1*2 + j]));
}

// u = Rz(w2) Ry(w1) Rx(w0) (per-qubit), U = kron(u, u)  -> Ure/Uim row-major 4x4
__device__ inline void build_unitary(const float* w, float* Ure, float* Uim) {
    const float h0 = w[0] * (W_MUL * 0.5f);
    const float h1 = w[1] * (W_MUL * 0.5f);
    const float h2 = w[2] * (W_MUL * 0.5f);
    const float c0 = cosf(h0), s0 = sinf(h0);
    const float c1 = cosf(h1), s1 = sinf(h1);
    const float cz = cosf(h2), sz = sinf(h2);
    C rx[4] = { {c0, 0.f}, {0.f, -s0}, {0.f, -s0}, {c0, 0.f} };
    C ry[4] = { {c1, 0.f}, {-s1, 0.f}, {s1, 0.f}, {c1, 0.f} };
    C rz[4] = { {cz, -sz}, {0.f, 0.f}, {0.f, 0.f}, {cz, sz} };
    C m[4], u[4];
    mm2(ry, rx, m);
    mm2(rz, m, u);
    #pragma unroll
    for (int a = 0; a < 2; ++a)
        #pragma unroll
        for (int bq = 0; bq < 2; ++bq)
            #pragma unroll
            for (int c = 0; c < 2; ++c)
                #pragma unroll
                for (int dq = 0; dq < 2; ++dq) {
                    C v = cmul(u[a*2 + c], u[bq*2 + dq]);
                    const int row = 2*a + bq, col = 2*c + dq;
                    Ure[row*4 + col] = v.r;
                    Uim[row*4 + col] = v.i;
                }
}

// Scalar recompute of w_hat(edge g) for the block-boundary inner product.
__device__ inline void edge_state(long long g,
                                  const long long* __restrict__ edges,
                                  const float2* __restrict__ nodef,
                                  const float* __restrict__ Ure,
                                  const float* __restrict__ Uim,
                                  float* wr, float* wi) {
    const long long n0 = edges[2*(size_t)g + 0];
    const long long n1 = edges[2*(size_t)g + 1];
    const float2 f0 = nodef[n0];
    const float2 f1 = nodef[n1];
    const float c[4] = { f0.x, f0.y, f1.x, f1.y };
    const float inv = rsqrtf(c[0]*c[0] + c[1]*c[1] + c[2]*c[2] + c[3]*c[3]);
    #pragma unroll
    for (int j = 0; j < 4; ++j) {
        float vr = 0.f, vi = 0.f;
        #pragma unroll
        for (int k = 0; k < 4; ++k) {
            vr += Ure[j*4 + k] * c[k];
            vi += Uim[j*4 + k] * c[k];
        }
        wr[j] = vr * inv;
        wi[j] = vi * inv;
    }
}

__global__ void __launch_bounds__(NTHREADS)
qugcn_edge_kernel(const long long* __restrict__ edges,
                  const float2*    __restrict__ nodef,
                  const float*     __restrict__ weight,
                  float*           __restrict__ ws)
{
    __shared__ float sUre[16], sUim[16];
    __shared__ float sWv[CHUNK * 8];      // unnormalized V: cols 0-3 re, 4-7 im
    __shared__ float sInv[CHUNK];         // 1/||c_e||
    __shared__ float sRed[2 * NTHREADS];  // complex product reduction

    const int t = threadIdx.x;
    const int chunkStart = blockIdx.x * CHUNK;

    if (t == 0) build_unitary(weight, sUre, sUim);
    __syncthreads();

    const unsigned lane = (unsigned)t & 31u;
    const unsigned half = lane >> 4;      // 0: K0/K1 & rows 0..7 ; 1: K2/K3 & rows 8..15
    const unsigned l15  = lane & 15u;
    const int waveId = t >> 5;            // 0..7

    // --- B operand (4x16): B[k][n] = n<4 ? Ure[n][k] : n<8 ? Uim[n-4][k] : 0
    // VGPR0 = rows K=0 (lanes 0-15) / K=2 (lanes 16-31); VGPR1 = K=1 / K=3.
    v2f bmat;
    {
        const unsigned k0 = half * 2u;
        float b0 = 0.f, b1 = 0.f;
        if (l15 < 4u)      { b0 = sUre[l15*4 + k0];     b1 = sUre[l15*4 + k0 + 1]; }
        else if (l15 < 8u) { b0 = sUim[(l15-4)*4 + k0]; b1 = sUim[(l15-4)*4 + k0 + 1]; }
        bmat[0] = b0; bmat[1] = b1;
    }

    // --- Phase 1: WMMA batched matvec, 16 edges per wave per iteration -------
    #pragma unroll 1
    for (int it = 0; it < CHUNK / 128; ++it) {
        const int localBase = it * 128 + waveId * 16;
        const int e = chunkStart + localBase + (int)l15;

        // A operand (16x4): lane L<16 holds c[e][0],c[e][1]; lane L+16 holds c[e][2],c[e][3]
        const long long node = edges[2*(size_t)e + half];
        const float2 f = nodef[node];
        v2f amat; amat[0] = f.x; amat[1] = f.y;

        if (it + 1 < CHUNK / 128)
            __builtin_prefetch(&edges[2*(size_t)(e + 128)], 0, 0);

        // row norm: combine the two lane halves of this edge
        const float part = f.x*f.x + f.y*f.y;
        const float n2 = part + __shfl_xor(part, 16, 32);
        if (lane < 16u) sInv[localBase + (int)lane] = rsqrtf(n2);

        v8f d = {0.f, 0.f, 0.f, 0.f, 0.f, 0.f, 0.f, 0.f};
        d = __builtin_amdgcn_wmma_f32_16x16x4_f32(
                /*neg_a=*/false, amat, /*neg_b=*/false, bmat,
                /*c_mod=*/(short)0, d, /*reuse_a=*/false, /*reuse_b=*/false);

        // D: VGPR r holds row r (lanes 0-15) / row r+8 (lanes 16-31), col = lane&15.
        // Only cols 0..7 carry data (cols 8..15 multiplied zeros).
        if (l15 < 8u) {
            const int rowOff = (int)half * 8;
            #pragma unroll
            for (int r = 0; r < 8; ++r)
                sWv[(localBase + rowOff + r) * 8 + (int)l15] = d[r];
        }
    }
    __syncthreads();

    // --- Phase 2: neighbor inner products + complex product ------------------
    float pr = 1.f, pi = 0.f;
    #pragma unroll
    for (int rep = 0; rep < CHUNK / NTHREADS; ++rep) {
        const int il = t + rep * NTHREADS;
        const long long gi = (long long)chunkStart + il;
        if (gi <= (long long)(N_EDGES - 2)) {
            const float ia = sInv[il];
            float ar[4], ai[4], br[4], bi[4];
            #pragma unroll
            for (int j = 0; j < 4; ++j) {
                ar[j] = sWv[il*8 + j]     * ia;
                ai[j] = sWv[il*8 + 4 + j] * ia;
            }
            if (il < CHUNK - 1) {
                const float ib = sInv[il + 1];
                #pragma unroll
                for (int j = 0; j < 4; ++j) {
                    br[j] = sWv[(il+1)*8 + j]     * ib;
                    bi[j] = sWv[(il+1)*8 + 4 + j] * ib;
                }
            } else {
                edge_state(gi + 1, edges, nodef, sUre, sUim, br, bi);
            }
            // q = <w_i, w_{i+1}> = sum conj(a)*b
            float qr = 0.f, qi = 0.f;
            #pragma unroll
            for (int j = 0; j < 4; ++j) {
                qr += ar[j]*br[j] + ai[j]*bi[j];
                qi += ar[j]*bi[j] - ai[j]*br[j];
            }
            const float nr = pr*qr - pi*qi;
            const float ni = pr*qi + pi*qr;
            pr = nr; pi = ni;
        }
    }

    sRed[2*t]     = pr;
    sRed[2*t + 1] = pi;
    __syncthreads();
    for (int s = NTHREADS / 2; s > 0; s >>= 1) {
        if (t < s) {
            const float xr = sRed[2*t],       xi = sRed[2*t + 1];
            const float yr = sRed[2*(t + s)], yi = sRed[2*(t + s) + 1];
            sRed[2*t]     = xr*yr - xi*yi;
            sRed[2*t + 1] = xr*yi + xi*yr;
        }
        __syncthreads();
    }

    if (t == 0) {
        ws[2*blockIdx.x]     = sRed[0];
        ws[2*blockIdx.x + 1] = sRed[1];
        if (blockIdx.x == 0) {
            const float inv0 = sInv[0];
            #pragma unroll
            for (int c = 0; c < 8; ++c) ws[WS_W0 + c] = sWv[c] * inv0;
        }
        if (blockIdx.x == NBLOCKS - 1) {
            const float invL = sInv[CHUNK - 1];
            #pragma unroll
            for (int c = 0; c < 8; ++c) ws[WS_WL + c] = sWv[(CHUNK - 1)*8 + c] * invL;
        }
    }
}

__global__ void qugcn_finalize(const float* __restrict__ ws,
                               const float* __restrict__ lin_w,
                               const float* __restrict__ lin_b,
                               float* __restrict__ out)
{
    if (threadIdx.x != 0 || blockIdx.x != 0) return;

    float sr = 1.f, si = 0.f;
    for (int b = 0; b < NBLOCKS; ++b) {
        const float prb = ws[2*b], pib = ws[2*b + 1];
        const float nr = sr*prb - si*pib;
        const float ni = sr*pib + si*prb;
        sr = nr; si = ni;
    }

    // chain[a][d] = s * w0[a] * conj(wL[d]); x = Re(chain) flattened row-major
    float x[16];
    #pragma unroll
    for (int a = 0; a < 4; ++a) {
        const float w0r = ws[WS_W0 + a], w0i = ws[WS_W0 + 4 + a];
        const float tr = sr*w0r - si*w0i;
        const float ti = sr*w0i + si*w0r;
        #pragma unroll
        for (int d = 0; d < 4; ++d) {
            const float wlr = ws[WS_WL + d], wli = ws[WS_WL + 4 + d];
            x[a*4 + d] = tr*wlr + ti*wli;
        }
    }

    float l0 = lin_b[0], l1 = lin_b[1];
    #pragma unroll
    for (int k = 0; k < 16; ++k) {
        l0 += x[k] * lin_w[k];
        l1 += x[k] * lin_w[16 + k];
    }
    const float m  = fmaxf(l0, l1);
    const float e0 = expf(l0 - m), e1 = expf(l1 - m);
    const float inv = 1.f / (e0 + e1);
    out[0] = e0 * inv;
    out[1] = e1 * inv;
}

extern "C" void kernel_launch(void* const* d_in, const int* in_sizes, int n_in,
                              void* d_out, int out_size, void* d_ws, size_t ws_size,
                              hipStream_t stream) {
    const long long* edges  = (const long long*)d_in[0];   // [E,2] int64
    const float2*    nodef  = (const float2*)   d_in[1];   // [N,2] f32
    const float*     weight = (const float*)    d_in[2];   // [5]
    const float*     lin_w  = (const float*)    d_in[3];   // [2,16]
    const float*     lin_b  = (const float*)    d_in[4];   // [2]
    float* ws  = (float*)d_ws;
    float* out = (float*)d_out;

    qugcn_edge_kernel<<<NBLOCKS, NTHREADS, 0, stream>>>(edges, nodef, weight, ws);
    qugcn_finalize<<<1, 32, 0, stream>>>(ws, lin_w, lin_b, out);

    (void)in_sizes; (void)n_in; (void)out_size; (void)ws_size;
}